// cooperative_localization_solver_38079180046907
// MI455X (gfx1250) — compile-verified
//
#include <hip/hip_runtime.h>

// ---------------------------------------------------------------------------
// Cooperative localization solver for MI455X (gfx1250), wave32.
//
// Dead code vs reference: the GRU branch (gru_*, m_sum) never reaches the
// returned outputs -> skipped entirely.
//
// CDNA5-specific paths used:
//  * V_WMMA_F32_16X16X4_F32 for the two ~0.5 GFLOP fp32 GEMMs
//      (E x 1500) w  @ (1500 x 32pad) we1[40:]
//      (E x 20) ubuf @ (20 x 1500)    u_w2
//  * GLOBAL_LOAD_ASYNC_TO_LDS_B128 (+ s_wait_asynccnt) staging the shared,
//    pre-swizzled WMMA-ready B panel into double-buffered LDS chunks,
//    overlapping async DMA with the WMMA inner loop; VMEM pipe reserved for
//    the 49MB streaming A operand.
//  * global_prefetch_b8 on the A stream.
// Workload is memory-bound (~300MB -> ~15us at 23.3 TB/s HBM).
// ---------------------------------------------------------------------------

typedef float v2f __attribute__((ext_vector_type(2)));
typedef float v8f __attribute__((ext_vector_type(8)));

constexpr int   N   = 1024;
constexpr int   D   = 4;
constexpr int   P   = 1500;
constexpr int   A   = 16;
constexpr int   E   = 8192;
constexpr int   DN  = 20;
constexpr int   DE  = 20;
constexpr int   DIN = 2 * DN + P;          // 1540
constexpr int   NSTEP = P / 4;             // 375 WMMA k-steps over the w panel
constexpr int   CK  = 300;                 // K-values per LDS chunk (75 steps, 5 chunks)
constexpr int   CSTEP = CK / 4;            // 75
constexpr int   CHUNK_FLOATS = CSTEP * 128; // 9600 floats = 38.4KB per buffer
constexpr float INV2S2 = 0.5f;             // 1/(2*sigma^2), sigma=1
constexpr float EPSF   = 1.1920928955078125e-07f;

__device__ __forceinline__ float leakyf(float x) { return x >= 0.f ? x : 0.01f * x; }
__device__ __forceinline__ float seluf(float x) {
    const float sc = 1.0507009873554805f, al = 1.6732632423543772f;
    return x > 0.f ? sc * x : sc * al * (expf(x) - 1.f);
}
__device__ __forceinline__ float sigmoidf_(float x) { return 1.f / (1.f + expf(-x)); }

// LDS byte offset of a __shared__ object: flat LDS addresses map to the LDS
// aperture whose low 32 bits are the in-LDS byte offset (ISA 10.2 aperture
// table).
__device__ __forceinline__ unsigned lds_off(const void* p) {
    return (unsigned)(unsigned long long)p;
}

// Async 16B copy global -> LDS, tracked by ASYNCcnt.
__device__ __forceinline__ void async_copy_b128(unsigned ldsoff, const void* gaddr) {
    asm volatile("global_load_async_to_lds_b128 %0, %1, off"
                 :: "v"(ldsoff), "v"(gaddr)
                 : "memory");
}
__device__ __forceinline__ void wait_async0() {
    asm volatile("s_wait_asynccnt 0x0" ::: "memory");
}

// -------- K_prep: pack we1[40:,:] into WMMA-ready swizzled B panel ---------
// we1q[((s*2 + T)*32 + lane)*2 + x] = B[4s + 2*kh + x][16T + n]   (0 if col>=20)
// where lane = kh*16 + n, B[k][j] = we1[(40+k)*20 + j].
// Each lane's (b.x,b.y) pair is then 8B-contiguous in LDS -> one ds_load_b64
// per WMMA B operand, conflict-free, no permute movs.
__global__ void k_pack_we1q(const float* __restrict__ we1, float* __restrict__ we1q) {
    int idx = blockIdx.x * 256 + threadIdx.x;
    if (idx < NSTEP * 2 * 32 * 2) {
        int x    = idx & 1;
        int lane = (idx >> 1) & 31;
        int T    = (idx >> 6) & 1;
        int s    = idx >> 7;
        int kh = lane >> 4, n = lane & 15;
        int k = 4 * s + 2 * kh + x;
        int j = 16 * T + n;
        we1q[idx] = (j < DE) ? we1[(2 * DN + k) * DE + j] : 0.f;
    }
}

// -------- K0: per-node mean + covariance -> h (N x 20) ---------------------
__global__ void k_node_stats(const float* __restrict__ parts, float* __restrict__ h_buf) {
    __shared__ float red[256];
    __shared__ float out[14];
    int n = blockIdx.x, t = threadIdx.x;
    const float* base = parts + (size_t)n * D * P;
    float s[4] = {0, 0, 0, 0};
    float sp[10];
    for (int i = 0; i < 10; i++) sp[i] = 0.f;
    for (int p = t; p < P; p += 256) {
        float v[4];
        for (int d = 0; d < D; d++) v[d] = base[d * P + p];
        int q = 0;
        for (int i = 0; i < D; i++) {
            s[i] += v[i];
            for (int j = i; j < D; j++) sp[q++] += v[i] * v[j];
        }
    }
    float vals[14];
    for (int i = 0; i < 4; i++) vals[i] = s[i];
    for (int i = 0; i < 10; i++) vals[4 + i] = sp[i];
    for (int k = 0; k < 14; k++) {
        red[t] = vals[k]; __syncthreads();
        for (int o = 128; o > 0; o >>= 1) { if (t < o) red[t] += red[t + o]; __syncthreads(); }
        if (t == 0) out[k] = red[0];
        __syncthreads();
    }
    if (t == 0) {
        float m[4];
        for (int d = 0; d < D; d++) m[d] = out[d] / (float)P;
        float* h = h_buf + n * DN;
        for (int d = 0; d < D; d++) h[d] = m[d];
        float cov[4][4];
        int q = 0;
        for (int i = 0; i < D; i++)
            for (int j = i; j < D; j++) {
                float c = out[4 + q] / (float)P - m[i] * m[j];
                cov[i][j] = c; cov[j][i] = c; q++;
            }
        for (int i = 0; i < D; i++)
            for (int j = 0; j < D; j++) h[4 + i * 4 + j] = cov[i][j];
    }
}

// -------- K1: anchor log-weights wa_log (N x P), row-max subtracted --------
__global__ void k_wa_log(const float* __restrict__ parts, const float* __restrict__ anchor_pos,
                         const float* __restrict__ meas, const float* __restrict__ mask,
                         float* __restrict__ wa_log) {
    __shared__ float ax[A], ay[A], am[A], ak[A];
    __shared__ float red[256];
    int n = blockIdx.x, t = threadIdx.x;
    if (t < A) {
        ax[t] = anchor_pos[(n * A + t) * 2 + 0];
        ay[t] = anchor_pos[(n * A + t) * 2 + 1];
        am[t] = meas[n * A + t];
        ak[t] = mask[n * A + t];
    }
    __syncthreads();
    const float* px = parts + (size_t)n * D * P;
    const float* py = px + P;
    float* wrow = wa_log + (size_t)n * P;
    float lmax = -3.4e38f;
    for (int p = t; p < P; p += 256) {
        float x = px[p], y = py[p], acc = 0.f;
        for (int a2 = 0; a2 < A; a2++) {
            float dx = x - ax[a2], dy = y - ay[a2];
            float pr = sqrtf(dx * dx + dy * dy);
            float dd = pr - am[a2];
            acc += -(dd * dd) * INV2S2 * ak[a2];
        }
        wrow[p] = acc;
        lmax = fmaxf(lmax, acc);
    }
    red[t] = lmax; __syncthreads();
    for (int o = 128; o > 0; o >>= 1) { if (t < o) red[t] = fmaxf(red[t], red[t + o]); __syncthreads(); }
    float m = red[0];
    for (int p = t; p < P; p += 256) wrow[p] -= m;
}

// -------- K2: edge particle weights w (E x P) ------------------------------
__global__ void k_edge_w(const float* __restrict__ parts, const float* __restrict__ edge_meas,
                         const int* __restrict__ esrc, const int* __restrict__ edst,
                         float* __restrict__ w_buf) {
    int e = blockIdx.x, t = threadIdx.x;
    int s = esrc[e], d = edst[e];
    float me = edge_meas[e];
    const float* sx = parts + (size_t)s * D * P;
    const float* sy = sx + P;
    const float* dx = parts + (size_t)d * D * P;
    const float* dy = dx + P;
    float* wrow = w_buf + (size_t)e * P;
    for (int p = t; p < P; p += 256) {
        float ddx = dx[p] - sx[p], ddy = dy[p] - sy[p];
        float pr = sqrtf(ddx * ddx + ddy * ddy);
        float df = pr - me;
        wrow[p] = expf(-(df * df) * INV2S2);
    }
}

// -------- K3: layer-1 GEMM, WMMA f32 16x16x4, async double-buffered B ------
// t1[e,j] = leaky( w[e,:] @ we1[40:,j] + h[dst]@we1[0:20,j] + h[src]@we1[20:40,j] + be1[j] )
// One wave per 16-edge tile; two 16-col C tiles. The pre-swizzled B panel
// (we1q, 187.5KB) is staged in 5 chunks of 38.4KB into 2 LDS buffers via
// GLOBAL_LOAD_ASYNC_TO_LDS_B128; chunk i+1's DMA overlaps chunk i's WMMAs.
// A-layout (16x4 f32): lanes 0-15 m=lane, K={0,1}; lanes 16-31 m=lane-16, K={2,3}.
__global__ void k_gemm1(const float* __restrict__ w_buf, const float* __restrict__ we1q,
                        const float* __restrict__ we1, const float* __restrict__ be1,
                        const float* __restrict__ h_buf, const int* __restrict__ esrc,
                        const int* __restrict__ edst, float* __restrict__ t1) {
    __shared__ __align__(16) float Bsh[2][CHUNK_FLOATS];   // 2 x 38.4 KB
    int t    = threadIdx.x;
    int lane = t & 31;
    int wv   = t >> 5;
    int e0   = (blockIdx.x * 8 + wv) * 16;
    int m    = lane & 15;
    int kh   = lane >> 4;

    v8f c0 = {0, 0, 0, 0, 0, 0, 0, 0};
    v8f c1 = c0;

    const float* arow = w_buf + (size_t)(e0 + m) * P + 2 * kh;     // 8B aligned

    // Prologue: stage chunk 0 into buffer 0.
    for (int i = t; i < CHUNK_FLOATS / 4; i += 256)
        async_copy_b128(lds_off(&Bsh[0][i * 4]), we1q + i * 4);
    wait_async0();
    __syncthreads();

    int cur = 0;
    for (int c = 0; c < P / CK; c++) {
        // Kick off next chunk's async DMA into the other buffer (safe: that
        // buffer's readers finished at the previous iteration's barrier).
        if (c + 1 < P / CK) {
            const float* gsrc = we1q + (size_t)(c + 1) * CHUNK_FLOATS;
            for (int i = t; i < CHUNK_FLOATS / 4; i += 256)
                async_copy_b128(lds_off(&Bsh[cur ^ 1][i * 4]), gsrc + i * 4);
        }

        // WMMA over current chunk: A streamed from global, B from LDS.
        const float* ar = arow + c * CK;
        const float* bb = &Bsh[cur][lane * 2];
        for (int s = 0; s < CSTEP; s++) {
            if ((s & 15) == 0) __builtin_prefetch(ar + 4 * s + 128, 0, 1); // global_prefetch_b8
            v2f a  = *(const v2f*)(ar + 4 * s);
            v2f b0 = *(const v2f*)(bb + s * 128);        // tile j = n
            v2f b1 = *(const v2f*)(bb + s * 128 + 64);   // tile j = 16 + n
            c0 = __builtin_amdgcn_wmma_f32_16x16x4_f32(false, a, false, b0, (short)0, c0, false, false);
            c1 = __builtin_amdgcn_wmma_f32_16x16x4_f32(false, a, false, b1, (short)0, c1, false, false);
        }

        wait_async0();       // next chunk's DMA done (overlapped with WMMAs)
        __syncthreads();     // all waves done reading cur before it is reused
        cur ^= 1;
    }

    // Epilogue: small K=40 h-contribution + bias + leaky, store t1 (stride 32).
    int n = m; // C/D layout: n = lane%16, row = r + 8*(lane>=16)
    for (int r = 0; r < 8; r++) {
        int mm = r + 8 * kh;
        int e  = e0 + mm;
        int ds = edst[e], ss = esrc[e];
        const float* hd = h_buf + ds * DN;
        const float* hs = h_buf + ss * DN;
        float a0 = c0[r], a1 = c1[r];
        for (int k = 0; k < DN; k++) {
            float dv = hd[k], sv = hs[k];
            a0 += dv * we1[k * DE + n] + sv * we1[(DN + k) * DE + n];
            if (n < DE - 16)
                a1 += dv * we1[k * DE + 16 + n] + sv * we1[(DN + k) * DE + 16 + n];
        }
        t1[(size_t)e * 32 + n] = leakyf(a0 + be1[n]);
        if (n < DE - 16)
            t1[(size_t)e * 32 + 16 + n] = leakyf(a1 + be1[16 + n]);
    }
}

// -------- K4: tiny per-edge MLPs (msg / sgate / ubuf), weights in LDS ------
__global__ void k_mlp(const float* __restrict__ t1,
                      const float* __restrict__ we2, const float* __restrict__ be2,
                      const float* __restrict__ s_w1, const float* __restrict__ s_b1,
                      const float* __restrict__ s_w2, const float* __restrict__ s_b2,
                      const float* __restrict__ u_w1, const float* __restrict__ u_b1,
                      float* __restrict__ sgate, float* __restrict__ ubuf) {
    __shared__ float Wwe2[400], Wsw1[400], Wuw1[400];
    __shared__ float Bbe2[20], Bsb1[20], Vsw2[20], Bub1[20];
    __shared__ float sb2s;
    for (int i = threadIdx.x; i < 400; i += 256) {
        Wwe2[i] = we2[i]; Wsw1[i] = s_w1[i]; Wuw1[i] = u_w1[i];
    }
    for (int i = threadIdx.x; i < 20; i += 256) {
        Bbe2[i] = be2[i]; Bsb1[i] = s_b1[i]; Vsw2[i] = s_w2[i]; Bub1[i] = u_b1[i];
    }
    if (threadIdx.x == 0) sb2s = s_b2[0];
    __syncthreads();

    int e = blockIdx.x * 256 + threadIdx.x;
    if (e >= E) return;
    float t[20], msg[20];
    for (int k = 0; k < 20; k++) t[k] = t1[(size_t)e * 32 + k];
    for (int j = 0; j < 20; j++) {
        float acc = Bbe2[j];
        for (int k = 0; k < 20; k++) acc += t[k] * Wwe2[k * 20 + j];
        msg[j] = leakyf(acc);
    }
    float sg = sb2s;
    for (int j = 0; j < 20; j++) {
        float acc = Bsb1[j];
        for (int k = 0; k < 20; k++) acc += msg[k] * Wsw1[k * 20 + j];
        sg += seluf(acc) * Vsw2[j];
    }
    sgate[e] = sigmoidf_(sg);
    for (int j = 0; j < 20; j++) {
        float acc = Bub1[j];
        for (int k = 0; k < 20; k++) acc += msg[k] * Wuw1[k * 20 + j];
        ubuf[(size_t)e * 32 + j] = seluf(acc);
    }
    for (int j = 20; j < 32; j++) ubuf[(size_t)e * 32 + j] = 0.f;
}

// -------- K5: uterm GEMM (WMMA, B tile async-staged in LDS) + w update -----
// All 8 waves of a block share the same 16-particle column tile -> stage the
// 20x16 u_w2 tile once per block via async-to-LDS (ragged last tile falls
// back to guarded scalar staging to avoid OOB global reads).
__global__ void k_uterm(const float* __restrict__ ubuf, const float* __restrict__ u_w2,
                        const float* __restrict__ u_b2, const float* __restrict__ sgate,
                        float* __restrict__ w_buf, int* __restrict__ maxw) {
    __shared__ __align__(16) float Bu[DE * 16];    // 1.28 KB
    int t    = threadIdx.x;
    int lane = t & 31;
    int wv   = t >> 5;
    int e0   = (blockIdx.x * 8 + wv) * 16;
    int p0   = blockIdx.y * 16;
    int m    = lane & 15;
    int kh   = lane >> 4;
    int n    = m;
    int pcol = p0 + n;
    bool pin = pcol < P;

    if (p0 + 16 <= P) {
        for (int i = t; i < DE * 4; i += 256) {    // 80 x B128 (row stride 6000B, 16B-mult)
            int row = i >> 2, seg = i & 3;
            async_copy_b128(lds_off(&Bu[row * 16 + seg * 4]),
                            u_w2 + (size_t)row * P + p0 + seg * 4);
        }
        wait_async0();
    } else {
        for (int i = t; i < DE * 16; i += 256) {
            int row = i >> 4, col = i & 15;
            Bu[i] = (p0 + col < P) ? u_w2[(size_t)row * P + p0 + col] : 0.f;
        }
    }
    __syncthreads();

    v8f c = {0, 0, 0, 0, 0, 0, 0, 0};
    const float* arow = ubuf + (size_t)(e0 + m) * 32 + 2 * kh;
#pragma unroll
    for (int k0 = 0; k0 < DE; k0 += 4) {
        v2f a = *(const v2f*)(arow + k0);
        int kr = k0 + 2 * kh;
        v2f b;
        b.x = Bu[kr * 16 + n];
        b.y = Bu[(kr + 1) * 16 + n];
        c = __builtin_amdgcn_wmma_f32_16x16x4_f32(false, a, false, b, (short)0, c, false, false);
    }
    float ub2v = pin ? u_b2[pcol] : 0.f;
    for (int r = 0; r < 8; r++) {
        int e = e0 + r + 8 * kh;
        float wnew = 0.f;
        if (pin) {
            float u = fmaxf(c[r] + ub2v, 0.f);
            size_t off = (size_t)e * P + pcol;
            wnew = sgate[e] * w_buf[off] + u;
            w_buf[off] = wnew;
        }
        // max over the 16 lanes sharing this edge row (all values >= 0)
        float mx = wnew;
        for (int o = 8; o > 0; o >>= 1) mx = fmaxf(mx, __shfl_xor(mx, o, 16));
        if (n == 0) atomicMax(maxw + e, __float_as_int(mx)); // nonneg float == int order
    }
}

// -------- K6: wl = log(w+eps) - log(max+eps), segment-sum into sum_wl ------
__global__ void k_segsum(const float* __restrict__ w_buf, const int* __restrict__ maxw,
                         const int* __restrict__ edst, float* __restrict__ sum_wl) {
    int e = blockIdx.x, t = threadIdx.x;
    float lmax = logf(__int_as_float(maxw[e]) + EPSF);
    int d = edst[e];
    const float* wrow = w_buf + (size_t)e * P;
    float* srow = sum_wl + (size_t)d * P;
    for (int p = t; p < P; p += 256)
        atomicAdd(srow + p, logf(wrow[p] + EPSF) - lmax);
}

// -------- K7: softmax belief, est_mean, est_cov ----------------------------
__global__ void k_belief(const float* __restrict__ wa_log, const float* __restrict__ sum_wl,
                         const float* __restrict__ parts, float* __restrict__ belief,
                         float* __restrict__ out_mean, float* __restrict__ out_cov) {
    __shared__ float red[256];
    __shared__ float sh[4];
    int n = blockIdx.x, t = threadIdx.x;
    const float* wrow = wa_log + (size_t)n * P;
    const float* srow = sum_wl + (size_t)n * P;
    float* brow = belief + (size_t)n * P;

    float lmax = -3.4e38f;
    for (int p = t; p < P; p += 256) lmax = fmaxf(lmax, wrow[p] + srow[p]);
    red[t] = lmax; __syncthreads();
    for (int o = 128; o > 0; o >>= 1) { if (t < o) red[t] = fmaxf(red[t], red[t + o]); __syncthreads(); }
    float M = red[0]; __syncthreads();

    float ssum = 0.f;
    for (int p = t; p < P; p += 256) { float ex = expf(wrow[p] + srow[p] - M); brow[p] = ex; ssum += ex; }
    red[t] = ssum; __syncthreads();
    for (int o = 128; o > 0; o >>= 1) { if (t < o) red[t] += red[t + o]; __syncthreads(); }
    float inv = 1.f / red[0]; __syncthreads();

    const float* pb = parts + (size_t)n * D * P;
    float macc[4] = {0, 0, 0, 0};
    for (int p = t; p < P; p += 256) {
        float b = brow[p] * inv;
        brow[p] = b;
        for (int d = 0; d < 4; d++) macc[d] += pb[d * P + p] * b;
    }
    for (int d = 0; d < 4; d++) {
        red[t] = macc[d]; __syncthreads();
        for (int o = 128; o > 0; o >>= 1) { if (t < o) red[t] += red[t + o]; __syncthreads(); }
        if (t == 0) { sh[d] = red[0]; out_mean[n * 4 + d] = red[0]; }
        __syncthreads();
    }
    float mean[4];
    for (int d = 0; d < 4; d++) mean[d] = sh[d];
    float cacc[10];
    for (int i = 0; i < 10; i++) cacc[i] = 0.f;
    for (int p = t; p < P; p += 256) {
        float b = brow[p];
        float v[4];
        for (int d = 0; d < 4; d++) v[d] = pb[d * P + p] - mean[d];
        int q = 0;
        for (int i = 0; i < 4; i++)
            for (int j = i; j < 4; j++) cacc[q++] += v[i] * v[j] * b;
    }
    int q = 0;
    for (int i = 0; i < 4; i++)
        for (int j = i; j < 4; j++) {
            red[t] = cacc[q]; __syncthreads();
            for (int o = 128; o > 0; o >>= 1) { if (t < o) red[t] += red[t + o]; __syncthreads(); }
            if (t == 0) {
                out_cov[n * 16 + i * 4 + j] = red[0];
                if (i != j) out_cov[n * 16 + j * 4 + i] = red[0];
            }
            __syncthreads();
            q++;
        }
}

// -------- K8: deterministic systematic resampling via CDF inversion --------
__global__ void k_resample(const float* __restrict__ belief, const float* __restrict__ parts,
                           float* __restrict__ out_res) {
    __shared__ float cdf[P];
    int n = blockIdx.x, t = threadIdx.x;
    const float* brow = belief + (size_t)n * P;
    for (int p = t; p < P; p += 256) cdf[p] = brow[p];
    __syncthreads();
    if (t == 0) {
        float run = 0.f;
        for (int p = 0; p < P; p++) { run += cdf[p]; cdf[p] = run; }
    }
    __syncthreads();
    float total = cdf[P - 1];
    const float* pb = parts + (size_t)n * D * P;
    float* ob = out_res + (size_t)n * D * P;
    for (int j = t; j < P; j += 256) {
        float u = (j + 0.5f) * total / (float)P;
        int lo = 0, hi = P - 1;
        while (lo < hi) { int mid = (lo + hi) >> 1; if (cdf[mid] < u) lo = mid + 1; else hi = mid; }
        for (int d = 0; d < 4; d++) ob[d * P + j] = pb[d * P + lo];
    }
}

// ---------------------------------------------------------------------------
extern "C" void kernel_launch(void* const* d_in, const int* in_sizes, int n_in,
                              void* d_out, int out_size, void* d_ws, size_t ws_size,
                              hipStream_t stream) {
    (void)in_sizes; (void)n_in; (void)out_size; (void)ws_size;
    const float* parts      = (const float*)d_in[0];
    const float* anchor_pos = (const float*)d_in[1];
    const float* meas       = (const float*)d_in[2];
    const float* mask       = (const float*)d_in[3];
    const float* edge_meas  = (const float*)d_in[4];
    const float* we1        = (const float*)d_in[5];
    const float* be1        = (const float*)d_in[6];
    const float* we2        = (const float*)d_in[7];
    const float* be2        = (const float*)d_in[8];
    // d_in[9..12]: GRU weights — dead code w.r.t. the returned outputs.
    const float* s_w1       = (const float*)d_in[13];
    const float* s_b1       = (const float*)d_in[14];
    const float* s_w2       = (const float*)d_in[15];
    const float* s_b2       = (const float*)d_in[16];
    const float* u_w1       = (const float*)d_in[17];
    const float* u_b1       = (const float*)d_in[18];
    const float* u_w2       = (const float*)d_in[19];
    const float* u_b2       = (const float*)d_in[20];
    const int*   esrc       = (const int*)d_in[21];
    const int*   edst       = (const int*)d_in[22];

    // Workspace carve-up (~70 MB of floats)
    float* ws = (float*)d_ws;
    size_t off = 0;
    float* w_buf  = ws + off; off += (size_t)E * P;     // 12.29M
    float* wa_log = ws + off; off += (size_t)N * P;     // 1.54M
    float* sum_wl = ws + off; off += (size_t)N * P;
    float* belief = ws + off; off += (size_t)N * P;
    float* h_buf  = ws + off; off += (size_t)N * DN;
    float* t1     = ws + off; off += (size_t)E * 32;
    float* ubuf   = ws + off; off += (size_t)E * 32;
    float* sgate  = ws + off; off += (size_t)E;
    int*   maxw   = (int*)(ws + off); off += (size_t)E;
    float* we1q   = ws + off; off += (size_t)NSTEP * 128;   // 48000 floats, 16B aligned

    float* out_mean = (float*)d_out;
    float* out_res  = out_mean + (size_t)N * D;
    float* out_cov  = out_res + (size_t)N * D * P;

    hipMemsetAsync(sum_wl, 0, (size_t)N * P * sizeof(float), stream);
    hipMemsetAsync(maxw, 0, (size_t)E * sizeof(int), stream);

    k_pack_we1q<<<(NSTEP * 128 + 255) / 256, 256, 0, stream>>>(we1, we1q);
    k_node_stats<<<N, 256, 0, stream>>>(parts, h_buf);
    k_wa_log<<<N, 256, 0, stream>>>(parts, anchor_pos, meas, mask, wa_log);
    k_edge_w<<<E, 256, 0, stream>>>(parts, edge_meas, esrc, edst, w_buf);
    k_gemm1<<<E / 16 / 8, 256, 0, stream>>>(w_buf, we1q, we1, be1, h_buf, esrc, edst, t1);
    k_mlp<<<E / 256, 256, 0, stream>>>(t1, we2, be2, s_w1, s_b1, s_w2, s_b2, u_w1, u_b1, sgate, ubuf);
    dim3 g5(E / 16 / 8, (P + 15) / 16);
    k_uterm<<<g5, 256, 0, stream>>>(ubuf, u_w2, u_b2, sgate, w_buf, maxw);
    k_segsum<<<E, 256, 0, stream>>>(w_buf, maxw, edst, sum_wl);
    k_belief<<<N, 256, 0, stream>>>(wa_log, sum_wl, parts, belief, out_mean, out_cov);
    k_resample<<<N, 256, 0, stream>>>(belief, parts, out_res);
}